// MSTAGATNetLite_1047972020696
// MI455X (gfx1250) — compile-verified
//
#include <hip/hip_runtime.h>
#include <hip/hip_bf16.h>
#include <math.h>

typedef __attribute__((ext_vector_type(2))) float v2f;
typedef __attribute__((ext_vector_type(8))) float v8f;

#define BATCH 2
#define T_LEN 64
#define NNODE 4096
#define CFE 16
#define BD 8
#define HID 32
#define NHEAD 4
#define HD 8
#define HOR 24
#define EPS_BN 1e-5f
#define EPS_LN 1e-5f

__device__ __forceinline__ float elu1(float x) { return x > 0.f ? x + 1.f : __expf(x); }

// 16-lane xor butterfly step via DPP16 ROW_XMASK (VALU, no LDS traffic)
template <int MASK>
__device__ __forceinline__ float row_xor(float v) {
  int t = __builtin_amdgcn_update_dpp(0, __float_as_int(v), 0x160 | MASK, 0xf, 0xf, true);
  return __int_as_float(t);
}
__device__ __forceinline__ float rmax16(float x) {
  x = fmaxf(x, row_xor<1>(x));
  x = fmaxf(x, row_xor<2>(x));
  x = fmaxf(x, row_xor<4>(x));
  x = fmaxf(x, row_xor<8>(x));
  return x;
}
__device__ __forceinline__ float rsum16(float x) {
  x += row_xor<1>(x);
  x += row_xor<2>(x);
  x += row_xor<4>(x);
  x += row_xor<8>(x);
  return x;
}

// ---------------------------------------------------------------------------
// Kernel 1: per-node temporal conv + BN + ReLU + p1 + p2 + qkv projection.
// Packs Q' = [(elu(q)+1)/sqrt(hd) | u_row]  (16 wide, bias folded into K-dim)
//       K' = [elu(k)+1            | v_col]  (16 wide)
//       V  = raw v, zero-padded to 16 wide (so attention V loads need no
//             predication; keeps EXEC all-1s through the WMMA loop)
// One 64-thread block per (b, n).
// ---------------------------------------------------------------------------
__global__ void fe_qkv_kernel(const float* __restrict__ x,
                              const float* __restrict__ fe_w, const float* __restrict__ fe_b,
                              const float* __restrict__ fe_bn_g, const float* __restrict__ fe_bn_b,
                              const float* __restrict__ p1_w, const float* __restrict__ p1_b,
                              const float* __restrict__ p2_w, const float* __restrict__ p2_b,
                              const float* __restrict__ qkv_w, const float* __restrict__ qkv_b,
                              const float* __restrict__ u, const float* __restrict__ v,
                              float* __restrict__ qp, float* __restrict__ kp,
                              float* __restrict__ vp) {
  const int bn = blockIdx.x;
  const int b = bn / NNODE, n = bn % NNODE;
  const int tid = threadIdx.x;  // 0..63 == time index
  __shared__ float xs[T_LEN + 2];
  __shared__ float f1[CFE * T_LEN];
  __shared__ float part[8 * 64];
  __shared__ float g1[8];
  __shared__ float g2[HID];

  xs[tid + 1] = x[((size_t)b * T_LEN + tid) * NNODE + n];
  if (tid == 0) { xs[0] = 0.f; xs[T_LEN + 1] = 0.f; }
  __syncthreads();

  const float bnscale = rsqrtf(1.f + EPS_BN);
  #pragma unroll
  for (int c = 0; c < CFE; ++c) {
    float o = fe_w[c * 3 + 0] * xs[tid] + fe_w[c * 3 + 1] * xs[tid + 1] +
              fe_w[c * 3 + 2] * xs[tid + 2] + fe_b[c];
    o = o * (fe_bn_g[c] * bnscale) + fe_bn_b[c];
    f1[c * T_LEN + tid] = fmaxf(o, 0.f);
  }
  __syncthreads();

  // p1: [1024] -> [8]
  float acc[8] = {0.f, 0.f, 0.f, 0.f, 0.f, 0.f, 0.f, 0.f};
  for (int i = tid; i < CFE * T_LEN; i += 64) {
    float fv = f1[i];
    #pragma unroll
    for (int j = 0; j < 8; ++j) acc[j] += fv * p1_w[j * (CFE * T_LEN) + i];
  }
  #pragma unroll
  for (int j = 0; j < 8; ++j) part[j * 64 + tid] = acc[j];
  __syncthreads();
  if (tid < 8) {
    float s = p1_b[tid];
    for (int t2 = 0; t2 < 64; ++t2) s += part[tid * 64 + t2];
    g1[tid] = fmaxf(s, 0.f);
  }
  __syncthreads();
  // p2: [8] -> [32]
  if (tid < HID) {
    float s = p2_b[tid];
    #pragma unroll
    for (int j = 0; j < 8; ++j) s += g1[j] * p2_w[tid * 8 + j];
    g2[tid] = s;
  }
  __syncthreads();

  const float qscale = 0.35355339059327373f;  // 1/sqrt(hd)
  for (int o = tid; o < 3 * HID; o += 64) {
    float s = qkv_b[o];
    #pragma unroll
    for (int c = 0; c < HID; ++c) s += g2[c] * qkv_w[o * HID + c];
    int sel = o / HID;
    int hh = (o % HID) / HD;
    int d = o % HD;
    size_t base = ((size_t)(b * NHEAD + hh) * NNODE + n);
    if (sel == 0)       qp[base * 16 + d] = elu1(s) * qscale;
    else if (sel == 1)  kp[base * 16 + d] = elu1(s);
    else                vp[base * 16 + d] = s;
  }
  // fold low-rank bias u@v into the extra 8 K-dim slots (same for every head)
  {
    int hh = tid >> 4, rem = tid & 15, i = rem & 7;
    size_t base = ((size_t)(b * NHEAD + hh) * NNODE + n) * 16 + 8 + i;
    if (rem < 8) qp[base] = u[(size_t)n * BD + i];
    else         kp[base] = v[(size_t)i * NNODE + n];
  }
  // zero-pad V columns 8..15 (removes predication from the attention loop)
  if (tid < 32) {
    int hh = tid >> 3, d = tid & 7;
    vp[((size_t)(b * NHEAD + hh) * NNODE + n) * 16 + 8 + d] = 0.f;
  }
}

// ---------------------------------------------------------------------------
// Kernel 2: split-K flash attention with V_WMMA_F32_16X16X4_F32.
// 4 waves per block, each streams 1/4 of the 256 column tiles with private
// online-softmax state (DPP ROW_XMASK reductions); partials merged in LDS
// with a log-sum-exp combine. 8 WMMAs per iteration; no predication inside
// the loop (EXEC stays all-1s).
// ---------------------------------------------------------------------------
#define KSPLIT 4
__global__ void __launch_bounds__(32 * KSPLIT)
attn_kernel(const float* __restrict__ qp, const float* __restrict__ kp,
            const float* __restrict__ vp, float* __restrict__ attn_out) {
  const int NT = NNODE / 16;          // 256 column tiles
  const int JW = NT / KSPLIT;         // 64 tiles per wave
  const int bh = blockIdx.x / NT;
  const int tile = blockIdx.x % NT;
  const int wid = threadIdx.x >> 5;
  const int lane = threadIdx.x & 31;
  const int m = lane & 15, half = lane >> 4;
  const int row0 = tile * 16;

  __shared__ float pl[KSPLIT][16 * 18];   // P transpose tile (stride 18: b64-aligned, conflict-free)
  __shared__ float mmax[KSPLIT][16];
  __shared__ float msum[KSPLIT][16];
  __shared__ float mo[KSPLIT][16][HD];
  float* plw = &pl[wid][0];

  // Q' tile in A layout (16x16 over 4 chunks of K=4); pairs are 8B-aligned
  const float* qrow = qp + ((size_t)bh * NNODE + row0 + m) * 16;
  v2f a[4];
  #pragma unroll
  for (int c = 0; c < 4; ++c) a[c] = *(const v2f*)(qrow + 4 * c + 2 * half);

  v8f o = {0.f, 0.f, 0.f, 0.f, 0.f, 0.f, 0.f, 0.f};
  float rmax[8], rsum[8];
  #pragma unroll
  for (int r = 0; r < 8; ++r) { rmax[r] = -3.0e38f; rsum[r] = 0.f; }

  const int j0 = wid * JW;
  for (int j = j0; j < j0 + JW; ++j) {
    const int col0 = j * 16;
    // K' tile in B layout
    const float* krow = kp + ((size_t)bh * NNODE + col0 + m) * 16;
    v2f bb[4];
    #pragma unroll
    for (int c = 0; c < 4; ++c) bb[c] = *(const v2f*)(krow + 4 * c + 2 * half);
    // V tile in B layout (V pre-padded to 16 cols: unconditional loads)
    v2f vb[4];
    #pragma unroll
    for (int c = 0; c < 4; ++c) {
      int k0 = col0 + 4 * c + 2 * half;
      vb[c].x = vp[((size_t)bh * NNODE + k0) * 16 + m];
      vb[c].y = vp[((size_t)bh * NNODE + k0 + 1) * 16 + m];
    }
    v8f s = {0.f, 0.f, 0.f, 0.f, 0.f, 0.f, 0.f, 0.f};
    #pragma unroll
    for (int c = 0; c < 4; ++c)
      s = __builtin_amdgcn_wmma_f32_16x16x4_f32(false, a[c], false, bb[c],
                                                (short)0, s, false, false);
    // online softmax: lane holds S[r + 8*half][m]; 16-lane DPP reductions
    #pragma unroll
    for (int r = 0; r < 8; ++r) {
      float sv = s[r];
      float nm = fmaxf(rmax[r], rmax16(sv));
      float cr = __expf(rmax[r] - nm);
      rmax[r] = nm;
      float p = __expf(sv - nm);
      rsum[r] = rsum[r] * cr + rsum16(p);
      o[r] *= cr;
      plw[(r + 8 * half) * 18 + m] = p;
    }
    // intra-wave LDS RAW fence (DS ops are in-order within a wave)
    asm volatile("s_wait_dscnt 0" ::: "memory");
    // P in A layout via LDS transpose
    v2f pa[4];
    #pragma unroll
    for (int c = 0; c < 4; ++c) pa[c] = *(const v2f*)(plw + m * 18 + 4 * c + 2 * half);
    #pragma unroll
    for (int c = 0; c < 4; ++c)
      o = __builtin_amdgcn_wmma_f32_16x16x4_f32(false, pa[c], false, vb[c],
                                                (short)0, o, false, false);
  }

  // publish per-wave partials
  if (m == 0) {
    #pragma unroll
    for (int r = 0; r < 8; ++r) {
      mmax[wid][r + 8 * half] = rmax[r];
      msum[wid][r + 8 * half] = rsum[r];
    }
  }
  if (m < HD) {
    #pragma unroll
    for (int r = 0; r < 8; ++r) mo[wid][r + 8 * half][m] = o[r];
  }
  __syncthreads();

  // wave 0 merges the 4 partials (log-sum-exp combine)
  if (threadIdx.x < 32 && m < HD) {
    const int b = bh / NHEAD, h = bh % NHEAD;
    #pragma unroll
    for (int r = 0; r < 8; ++r) {
      int row = r + 8 * half;
      float M = mmax[0][row];
      #pragma unroll
      for (int w = 1; w < KSPLIT; ++w) M = fmaxf(M, mmax[w][row]);
      float ssum = 0.f, oacc = 0.f;
      #pragma unroll
      for (int w = 0; w < KSPLIT; ++w) {
        float e = __expf(mmax[w][row] - M);
        ssum += msum[w][row] * e;
        oacc += mo[w][row][m] * e;
      }
      attn_out[((size_t)b * NNODE + row0 + row) * HID + h * HD + m] = oacc / ssum;
    }
  }
}

// ---------------------------------------------------------------------------
// Kernel 3: output projection s = attn_out @ op_w^T + op_b
// ---------------------------------------------------------------------------
__global__ void op_proj_kernel(const float* __restrict__ attn_out,
                               const float* __restrict__ op_w,
                               const float* __restrict__ op_b,
                               float* __restrict__ sbuf) {
  int idx = blockIdx.x * blockDim.x + threadIdx.x;
  if (idx >= BATCH * NNODE * HID) return;
  int c = idx & (HID - 1);
  int bn = idx / HID;
  const float* arow = attn_out + (size_t)bn * HID;
  float s = op_b[c];
  #pragma unroll
  for (int k2 = 0; k2 < HID; ++k2) s += arow[k2] * op_w[c * HID + k2];
  sbuf[idx] = s;
}

// ---------------------------------------------------------------------------
// Kernel 4: dilated node convs + BN + ReLU + residual LN + horizon MLP.
// 64-node tile (+2 halo each side) in LDS; one thread per node.
// ---------------------------------------------------------------------------
__global__ void tail_kernel(const float* __restrict__ sbuf,
                            const float* __restrict__ tc0_w, const float* __restrict__ tc0_b,
                            const float* __restrict__ tc1_w, const float* __restrict__ tc1_b,
                            const float* __restrict__ t_bn_g, const float* __restrict__ t_bn_b,
                            const float* __restrict__ t_ln_g, const float* __restrict__ t_ln_b,
                            const float* __restrict__ h1_w, const float* __restrict__ h1_b,
                            const float* __restrict__ h_ln_g, const float* __restrict__ h_ln_b,
                            const float* __restrict__ h2_w, const float* __restrict__ h2_b,
                            float* __restrict__ out) {
  const int TILE = 64;
  const int b = blockIdx.x / (NNODE / TILE);
  const int n0 = (blockIdx.x % (NNODE / TILE)) * TILE;
  const int tid = threadIdx.x;
  __shared__ float st[(TILE + 4) * HID];
  for (int i = tid; i < (TILE + 4) * HID; i += TILE) {
    int row = i / HID, c = i % HID;
    int nn = n0 - 2 + row;
    st[i] = (nn >= 0 && nn < NNODE) ? sbuf[((size_t)b * NNODE + nn) * HID + c] : 0.f;
  }
  __syncthreads();

  const int n = n0 + tid;
  const float* s0 = &st[(tid + 2) * HID];
  const float bnscale = rsqrtf(1.f + EPS_BN);
  float fused[HID];
  float mu = 0.f;
  for (int c = 0; c < HID; ++c) {
    float a0 = tc0_b[c], a1 = tc1_b[c];
    for (int cin = 0; cin < HID; ++cin) {
      const float* w0 = &tc0_w[(c * HID + cin) * 3];
      const float* w1 = &tc1_w[(c * HID + cin) * 3];
      float sm2 = st[(tid + 0) * HID + cin];
      float sm1 = st[(tid + 1) * HID + cin];
      float sc  = st[(tid + 2) * HID + cin];
      float sp1 = st[(tid + 3) * HID + cin];
      float sp2 = st[(tid + 4) * HID + cin];
      a0 += w0[0] * sm1 + w0[1] * sc + w0[2] * sp1;   // dilation 1, pad 1
      a1 += w1[0] * sm2 + w1[1] * sc + w1[2] * sp2;   // dilation 2, pad 2
    }
    float y0 = fmaxf(a0 * (t_bn_g[c] * bnscale) + t_bn_b[c], 0.f);
    float y1 = fmaxf(a1 * (t_bn_g[c] * bnscale) + t_bn_b[c], 0.f);
    float f = 0.5f * (y0 + y1) + s0[c];
    fused[c] = f;
    mu += f;
  }
  mu *= (1.f / HID);
  float var = 0.f;
  #pragma unroll
  for (int c = 0; c < HID; ++c) { float d = fused[c] - mu; var += d * d; }
  var *= (1.f / HID);
  float inv = rsqrtf(var + EPS_LN);
  float tout[HID];
  #pragma unroll
  for (int c = 0; c < HID; ++c) tout[c] = (fused[c] - mu) * inv * t_ln_g[c] + t_ln_b[c];

  float hh[BD];
  float mu2 = 0.f;
  #pragma unroll
  for (int j = 0; j < BD; ++j) {
    float s = h1_b[j];
    for (int c = 0; c < HID; ++c) s += tout[c] * h1_w[j * HID + c];
    hh[j] = s;
    mu2 += s;
  }
  mu2 *= (1.f / BD);
  float var2 = 0.f;
  #pragma unroll
  for (int j = 0; j < BD; ++j) { float d = hh[j] - mu2; var2 += d * d; }
  var2 *= (1.f / BD);
  float inv2 = rsqrtf(var2 + EPS_LN);
  #pragma unroll
  for (int j = 0; j < BD; ++j)
    hh[j] = fmaxf((hh[j] - mu2) * inv2 * h_ln_g[j] + h_ln_b[j], 0.f);

  #pragma unroll
  for (int ho = 0; ho < HOR; ++ho) {
    float s = h2_b[ho];
    #pragma unroll
    for (int j = 0; j < BD; ++j) s += hh[j] * h2_w[ho * BD + j];
    out[((size_t)b * HOR + ho) * NNODE + n] = s;  // [B, horizon, N]
  }
}

// ---------------------------------------------------------------------------
extern "C" void kernel_launch(void* const* d_in, const int* in_sizes, int n_in,
                              void* d_out, int out_size, void* d_ws, size_t ws_size,
                              hipStream_t stream) {
  const float* x       = (const float*)d_in[0];
  const float* fe_w    = (const float*)d_in[1];
  const float* fe_b    = (const float*)d_in[2];
  const float* fe_bn_g = (const float*)d_in[3];
  const float* fe_bn_b = (const float*)d_in[4];
  const float* p1_w    = (const float*)d_in[5];
  const float* p1_b    = (const float*)d_in[6];
  const float* p2_w    = (const float*)d_in[7];
  const float* p2_b    = (const float*)d_in[8];
  const float* qkv_w   = (const float*)d_in[9];
  const float* qkv_b   = (const float*)d_in[10];
  const float* op_w    = (const float*)d_in[11];
  const float* op_b    = (const float*)d_in[12];
  const float* u       = (const float*)d_in[13];
  const float* v       = (const float*)d_in[14];
  const float* tc0_w   = (const float*)d_in[15];
  const float* tc0_b   = (const float*)d_in[16];
  const float* tc1_w   = (const float*)d_in[17];
  const float* tc1_b   = (const float*)d_in[18];
  const float* t_bn_g  = (const float*)d_in[19];
  const float* t_bn_b  = (const float*)d_in[20];
  const float* t_ln_g  = (const float*)d_in[21];
  const float* t_ln_b  = (const float*)d_in[22];
  const float* h1_w    = (const float*)d_in[23];
  const float* h1_b    = (const float*)d_in[24];
  const float* h_ln_g  = (const float*)d_in[25];
  const float* h_ln_b  = (const float*)d_in[26];
  const float* h2_w    = (const float*)d_in[27];
  const float* h2_b    = (const float*)d_in[28];

  float* ws = (float*)d_ws;
  float* qp   = ws;                                          // [B,H,N,16]
  float* kp   = qp + (size_t)BATCH * NHEAD * NNODE * 16;     // [B,H,N,16]
  float* vp   = kp + (size_t)BATCH * NHEAD * NNODE * 16;     // [B,H,N,16] (padded)
  float* attn = vp + (size_t)BATCH * NHEAD * NNODE * 16;     // [B,N,32]
  float* sbuf = attn + (size_t)BATCH * NNODE * HID;          // [B,N,32]

  fe_qkv_kernel<<<BATCH * NNODE, 64, 0, stream>>>(
      x, fe_w, fe_b, fe_bn_g, fe_bn_b, p1_w, p1_b, p2_w, p2_b,
      qkv_w, qkv_b, u, v, qp, kp, vp);

  attn_kernel<<<BATCH * NHEAD * (NNODE / 16), 32 * KSPLIT, 0, stream>>>(qp, kp, vp, attn);

  op_proj_kernel<<<(BATCH * NNODE * HID + 255) / 256, 256, 0, stream>>>(
      attn, op_w, op_b, sbuf);

  tail_kernel<<<BATCH * (NNODE / 64), 64, 0, stream>>>(
      sbuf, tc0_w, tc0_b, tc1_w, tc1_b, t_bn_g, t_bn_b, t_ln_g, t_ln_b,
      h1_w, h1_b, h_ln_g, h_ln_b, h2_w, h2_b, (float*)d_out);
}